// DT_loss_42820823941428
// MI455X (gfx1250) — compile-verified
//
#include <hip/hip_runtime.h>
#include <hip/hip_bf16.h>

// Scene dims fixed by the reference setup.
constexpr int H = 704, W = 704, Dz = 64;
constexpr int BLK = 256;                 // 8 wave32s per workgroup
constexpr int WAVES = BLK / 32;

typedef __attribute__((ext_vector_type(2))) float f2;

__device__ __forceinline__ float wave_reduce_add(float v) {
    #pragma unroll
    for (int off = 16; off > 0; off >>= 1)
        v += __shfl_down(v, off, 32);
    return v;
}

// Kernel 1: per-point trilinear gather + deterministic per-block partial sum.
__global__ __launch_bounds__(BLK) void dt_gather_kernel(
    const float* __restrict__ pc1,
    const float* __restrict__ flow,
    const float* __restrict__ grid,
    const float* __restrict__ grid_min,
    const int*   __restrict__ grid_factor,
    float* __restrict__ out_dist,
    float* __restrict__ block_sums,
    int N)
{
    __shared__ float wsum[WAVES];
    const int tid = threadIdx.x;
    const int i   = blockIdx.x * BLK + tid;

    // Uniform (wave-scalar) parameters.
    const float gf  = (float)(*grid_factor);
    const float gmx = grid_min[0], gmy = grid_min[1], gmz = grid_min[2];

    float d = 0.0f;
    if (i < N) {
        // Speculative (DEV-scope) prefetch of the streaming input ~8 blocks
        // ahead; OOB translations are silently dropped on gfx1250.
        const int pf = 3 * (i + 8 * BLK);
        __builtin_prefetch(pc1  + pf, 0, 1);
        __builtin_prefetch(flow + pf, 0, 1);

        const int b = 3 * i;
        const float px = pc1[b + 0] + flow[b + 0];
        const float py = pc1[b + 1] + flow[b + 1];
        const float pz = pc1[b + 2] + flow[b + 2];

        float sx = fminf(fmaxf((px - gmx) * gf, 0.0f), (float)(H  - 1));
        float sy = fminf(fmaxf((py - gmy) * gf, 0.0f), (float)(W  - 1));
        float sz = fminf(fmaxf((pz - gmz) * gf, 0.0f), (float)(Dz - 1));

        const float fx = floorf(sx), fy = floorf(sy), fz = floorf(sz);
        const int x0 = (int)fx, y0 = (int)fy, z0 = (int)fz;
        const int x1 = min(x0 + 1, H - 1);
        const int y1 = min(y0 + 1, W - 1);
        const float wx = sx - fx, wy = sy - fy, wz = sz - fz;

        // z-pair packing: load float2 at zb = min(z0, Dz-2).
        //   z1 = min(z0+1, Dz-1) == zb+1 always  -> c_z1 = pair.y
        //   c_z0 = (z0 < Dz-1) ? pair.x : pair.y
        const int  zb  = min(z0, Dz - 2);
        const bool zin = (z0 < Dz - 1);

        // Linear row index (x*W + y)*Dz ; max ~31.7M fits in int32.
        const int r00 = (x0 * W + y0) * Dz + zb;
        const int r01 = (x0 * W + y1) * Dz + zb;
        const int r10 = (x1 * W + y0) * Dz + zb;
        const int r11 = (x1 * W + y1) * Dz + zb;

        // 4 b64 gathers (DWORD-aligned; legal in default alignment mode).
        // Grid (127 MB) is L2-resident on MI455X (192 MB L2).
        const f2 g00 = *(const f2*)(grid + r00);
        const f2 g01 = *(const f2*)(grid + r01);
        const f2 g10 = *(const f2*)(grid + r10);
        const f2 g11 = *(const f2*)(grid + r11);

        const float c000 = zin ? g00.x : g00.y, c001 = g00.y;
        const float c010 = zin ? g01.x : g01.y, c011 = g01.y;
        const float c100 = zin ? g10.x : g10.y, c101 = g10.y;
        const float c110 = zin ? g11.x : g11.y, c111 = g11.y;

        const float omz = 1.0f - wz, omy = 1.0f - wy, omx = 1.0f - wx;
        const float c00 = c000 * omz + c001 * wz;
        const float c01 = c010 * omz + c011 * wz;
        const float c10 = c100 * omz + c101 * wz;
        const float c11 = c110 * omz + c111 * wz;
        const float c0  = c00 * omy + c01 * wy;
        const float c1  = c10 * omy + c11 * wy;
        d = c0 * omx + c1 * wx;

        out_dist[i] = d;
    }

    // Deterministic reduction: fixed wave32 shuffle tree, then fixed-order
    // accumulation of the 8 wave partials by thread 0.
    const float ws = wave_reduce_add(d);
    if ((tid & 31) == 0) wsum[tid >> 5] = ws;
    __syncthreads();
    if (tid == 0) {
        float t = 0.0f;
        #pragma unroll
        for (int j = 0; j < WAVES; ++j) t += wsum[j];
        block_sums[blockIdx.x] = t;
    }
}

// Kernel 2: deterministic final reduction of block partial sums -> mean.
__global__ __launch_bounds__(BLK) void dt_mean_kernel(
    const float* __restrict__ block_sums, int nblocks,
    float* __restrict__ out_mean, float invN)
{
    __shared__ float wsum[WAVES];
    float s = 0.0f;
    for (int j = threadIdx.x; j < nblocks; j += BLK)  // fixed per-thread order
        s += block_sums[j];
    const float ws = wave_reduce_add(s);
    if ((threadIdx.x & 31) == 0) wsum[threadIdx.x >> 5] = ws;
    __syncthreads();
    if (threadIdx.x == 0) {
        float t = 0.0f;
        #pragma unroll
        for (int j = 0; j < WAVES; ++j) t += wsum[j];
        out_mean[0] = t * invN;
    }
}

extern "C" void kernel_launch(void* const* d_in, const int* in_sizes, int n_in,
                              void* d_out, int out_size, void* d_ws, size_t ws_size,
                              hipStream_t stream)
{
    const float* pc1         = (const float*)d_in[0];   // [1,N,3]
    const float* flow        = (const float*)d_in[1];   // [1,N,3]
    const float* dist_grid   = (const float*)d_in[2];   // [H,W,Dz]
    const float* grid_min    = (const float*)d_in[3];   // [3]
    const int*   grid_factor = (const int*)d_in[4];     // scalar

    const int N = in_sizes[0] / 3;
    const int nblocks = (N + BLK - 1) / BLK;

    float* out_mean   = (float*)d_out;      // d_out[0] = mean
    float* out_dist   = (float*)d_out + 1;  // d_out[1..N] = dist
    float* block_sums = (float*)d_ws;       // nblocks floats of scratch

    dt_gather_kernel<<<nblocks, BLK, 0, stream>>>(
        pc1, flow, dist_grid, grid_min, grid_factor,
        out_dist, block_sums, N);

    dt_mean_kernel<<<1, BLK, 0, stream>>>(
        block_sums, nblocks, out_mean, 1.0f / (float)N);
}